// locally_connected_1d_ch_1125281431913
// MI455X (gfx1250) — compile-verified
//
#include <hip/hip_runtime.h>

typedef __attribute__((ext_vector_type(2))) float v2f;
typedef __attribute__((ext_vector_type(8))) float v8f;

// out[b][p][o] = sum_c x[b][p][c] * w[p][c][o] + bias[p][o]
// B=32, C=64, OUT=4 hardcoded; P passed in.
// One wave per position p: two 16-batch M-tiles, N=16 (cols 0..3 valid),
// K=64 as 16 x V_WMMA_F32_16X16X4_F32 steps per tile.
__global__ __launch_bounds__(256) void lc1d_wmma_kernel(
    const float* __restrict__ x,     // [32][P][64]
    const float* __restrict__ w,     // [P][64][4]
    const float* __restrict__ bias,  // [P][4]
    float* __restrict__ out,         // [32][P][4]
    int P) {
  const int lane = threadIdx.x & 31;
  const int wave = threadIdx.x >> 5;
  const int p = blockIdx.x * 8 + wave;
  if (p >= P) return;  // wave-uniform: EXEC stays all-1s for live waves

  const int m     = lane & 15;        // A row / B-D column index within half
  const int khalf = (lane >> 4) * 2;  // 0 for lanes 0-15, 2 for lanes 16-31

  // ---- B fragment addressing (kernel weights), zero-pad columns n>=4 ----
  const int   n      = m;
  const bool  nvalid = (n < 4);
  const int   nc     = nvalid ? n : 3;              // clamped (safe) address
  const float* wp    = w + (size_t)p * 64 * 4;      // [c][o], o stride 1

  // ---- A fragment addressing (x), contiguous float2 per lane ----
  const size_t bstride = (size_t)P * 64;            // batch stride in elements
  const float* x0 = x + (size_t)m * bstride + (size_t)p * 64 + khalf; // b = m
  const float* x1 = x0 + 16 * bstride;                                // b = 16+m

  v8f acc0 = {};  // rows 0..15 of batch
  v8f acc1 = {};  // rows 16..31 of batch

#pragma unroll
  for (int kk = 0; kk < 64; kk += 4) {
    // A: lane holds A[m][kk+khalf+0], A[m][kk+khalf+1]  (global_load_b64)
    v2f a0 = *(const v2f*)(x0 + kk);
    v2f a1 = *(const v2f*)(x1 + kk);

    // B: lane holds B[kk+khalf+0][n], B[kk+khalf+1][n]; zero for n>=4
    float b0 = wp[(kk + khalf + 0) * 4 + nc];
    float b1 = wp[(kk + khalf + 1) * 4 + nc];
    v2f bf;
    bf.x = nvalid ? b0 : 0.0f;   // v_cndmask, no EXEC change
    bf.y = nvalid ? b1 : 0.0f;

    // D = A x B + C   (8 args: neg_a, A, neg_b, B, c_mod, C, reuse_a, reuse_b)
    acc0 = __builtin_amdgcn_wmma_f32_16x16x4_f32(
        false, a0, false, bf, (short)0, acc0, false, false);
    acc1 = __builtin_amdgcn_wmma_f32_16x16x4_f32(
        false, a1, false, bf, (short)0, acc1, false, false);
  }

  // ---- Store: D layout — lanes 0-15 hold rows 0..7 (vgpr r -> M=r),
  //      lanes 16-31 hold rows 8..15 (vgpr r -> M=8+r); N = lane%16.
  if (nvalid) {
    const float bv    = bias[(size_t)p * 4 + n];
    const int   mbase = (lane >> 4) * 8;
    float* o0 = out + (size_t)p * 4 + n;
    const size_t ostride = (size_t)P * 4;  // batch stride of out
#pragma unroll
    for (int r = 0; r < 8; ++r) {
      const int row = mbase + r;
      o0[(size_t)row * ostride]        = acc0[r] + bv;  // b = row
      o0[(size_t)(row + 16) * ostride] = acc1[r] + bv;  // b = row + 16
    }
  }
}

extern "C" void kernel_launch(void* const* d_in, const int* in_sizes, int n_in,
                              void* d_out, int out_size, void* d_ws, size_t ws_size,
                              hipStream_t stream) {
  const float* x    = (const float*)d_in[0];  // [32][P][64]
  const float* w    = (const float*)d_in[1];  // [1][P][64][4]
  const float* bias = (const float*)d_in[2];  // [P][4]
  float*       out  = (float*)d_out;          // [32][P][4]

  const int P = in_sizes[2] / 4;  // bias has P*4 elements

  dim3 block(256);                 // 8 waves (wave32), one position each
  dim3 grid((P + 7) / 8);
  lc1d_wmma_kernel<<<grid, block, 0, stream>>>(x, w, bias, out, P);
}